// IGN2to1_10900626997816
// MI455X (gfx1250) — compile-verified
//
#include <hip/hip_runtime.h>
#include <cstddef>

typedef __attribute__((ext_vector_type(2))) float v2f;
typedef __attribute__((ext_vector_type(8))) float v8f;

// ---------------------------------------------------------------- utilities
__device__ __forceinline__ float wave32_sum(float v) {
#pragma unroll
  for (int m = 16; m >= 1; m >>= 1) v += __shfl_xor(v, m, 32);
  return v;
}
// butterfly within each 16-lane half of a wave32 (masks < 16 stay in-half)
__device__ __forceinline__ float half16_sum(float v) {
#pragma unroll
  for (int m = 1; m < 16; m <<= 1) v += __shfl_xor(v, m, 32);
  return v;
}

__device__ __forceinline__ v8f wmma_f32_k4(v2f a, v2f b, v8f c) {
  // V_WMMA_F32_16X16X4_F32: D = A(16x4) * B(4x16) + C(16x16), all fp32
  return __builtin_amdgcn_wmma_f32_16x16x4_f32(false, a, false, b, (short)0, c,
                                               false, false);
}

// ============================================================== K0: zero
__global__ void k_zero(float* p, int n) {
  int i = blockIdx.x * blockDim.x + threadIdx.x;
  if (i < n) p[i] = 0.f;
}

// ============================================================== K1: reduce x
// 512 blocks: one per (matrix, row-block of 128). 512 threads = 16 waves.
// Wave w handles rows {r0+w, r0+w+16, ...} (8 rows). Lane l reads float4 at
// cols 128c+4l (fully coalesced b128). Column/diagsum/total partials are
// combined across row-blocks with global f32 atomics (L2-resident).
__global__ void k_reduce_x(const float* __restrict__ x,
                           float* __restrict__ rowsO, float* __restrict__ colsO,
                           float* __restrict__ diagO, float* __restrict__ dsumO,
                           float* __restrict__ totO) {
  __shared__ float colbuf[16 * 512];  // 32 KB: per-wave column partials
  __shared__ float totbuf[16];

  const int bid = blockIdx.x;
  const int mat = bid >> 2;           // (n*16 + d)
  const int r0 = (bid & 3) << 7;      // row-block base
  const int tid = threadIdx.x;
  const int w = tid >> 5, l = tid & 31;
  const float* M = x + (size_t)mat * 262144;  // 512*512

  float ca[16];
#pragma unroll
  for (int i = 0; i < 16; ++i) ca[i] = 0.f;
  float totp = 0.f;

#pragma unroll 4  // keep ~16 b128 loads in flight per thread (MLP for HBM)
  for (int t = 0; t < 8; ++t) {
    const int r = r0 + w + (t << 4);
    const float4* rp = (const float4*)(M + (size_t)r * 512);
    float rsum = 0.f;
#pragma unroll
    for (int c = 0; c < 4; ++c) {
      float4 v = rp[(c << 5) + l];  // cols 128c + 4l .. +3
      ca[c * 4 + 0] += v.x; ca[c * 4 + 1] += v.y;
      ca[c * 4 + 2] += v.z; ca[c * 4 + 3] += v.w;
      rsum += (v.x + v.y) + (v.z + v.w);
    }
    rsum = wave32_sum(rsum);
    if (l == 0) { rowsO[mat * 512 + r] = rsum; totp += rsum; }
  }
  // dump per-wave column partials to LDS
#pragma unroll
  for (int c = 0; c < 4; ++c)
#pragma unroll
    for (int j = 0; j < 4; ++j)
      colbuf[w * 512 + c * 128 + l * 4 + j] = ca[c * 4 + j];
  if (l == 0) totbuf[w] = totp;

  // diagonal slice owned by this row-block (threads 0..127)
  float dv = 0.f;
  if (tid < 128) {
    const int r = r0 + tid;
    dv = M[(size_t)r * 512 + r];
    diagO[mat * 512 + r] = dv;
  }
  float ds = wave32_sum(dv);

  __syncthreads();
  float csum = 0.f;
#pragma unroll
  for (int w2 = 0; w2 < 16; ++w2) csum += colbuf[w2 * 512 + tid];
  atomicAdd(&colsO[mat * 512 + tid], csum);

  if (l == 0 && w < 4) atomicAdd(&dsumO[mat], ds);
  if (tid == 0) {
    float T = 0.f;
#pragma unroll
    for (int i = 0; i < 16; ++i) T += totbuf[i];
    atomicAdd(&totO[mat], T);
  }
}

// ===================================== K1b: build layer-2 matrices + constants
// W0[s*16+d]=c2[d,s,0]; W2=c2[d,s,2]/512; W3=c2[d,s,3]/512
// cst0[n*64+s]=bias2[s]+sum_d(c2[d,s,1]*dsum[n,d]/512 + c2[d,s,4]*tot[n,d]/512^2)
__global__ void k_prep2(const float* __restrict__ c2, const float* __restrict__ bias2,
                        const float* __restrict__ dsum, const float* __restrict__ tot,
                        float* __restrict__ W0, float* __restrict__ W2,
                        float* __restrict__ W3, float* __restrict__ cst0) {
  const int tid = threadIdx.x;
  for (int idx = tid; idx < 1024; idx += 256) {
    int s = idx >> 4, d = idx & 15;
    const float* c = c2 + (size_t)(d * 64 + s) * 5;
    W0[idx] = c[0];
    W2[idx] = c[2] * (1.f / 512.f);
    W3[idx] = c[3] * (1.f / 512.f);
  }
  for (int idx = tid; idx < 512; idx += 256) {
    int n = idx >> 6, s = idx & 63;
    float acc = bias2[s];
    for (int d = 0; d < 16; ++d) {
      const float* c = c2 + (size_t)(d * 64 + s) * 5;
      acc += c[1] * dsum[n * 16 + d] * (1.f / 512.f) +
             c[4] * tot[n * 16 + d] * (1.f / (512.f * 512.f));
    }
    cst0[idx] = acc;
  }
}

// ================================ helper: K=16 WMMA GEMM slice (A 64x16 packed)
__device__ __forceinline__ v8f gemm_k16(v8f acc, const float* __restrict__ W,
                                        const float* __restrict__ S, int s0,
                                        int i0, int q, int h) {
#pragma unroll
  for (int k0 = 0; k0 < 16; k0 += 4) {
    const int ka = k0 + 2 * h;
    v2f a, b;
    a.x = W[(s0 + q) * 16 + ka];
    a.y = W[(s0 + q) * 16 + ka + 1];
    b.x = S[ka * 512 + i0 + q];
    b.y = S[(ka + 1) * 512 + i0 + q];
    acc = wmma_f32_k4(a, b, acc);
  }
  return acc;
}

// ============================================= K2: layer_2_to_1 + ReLU + stats
// out[n,s,i] = relu( W0@diag + W2@rows + W3@cols + cst0[n,s] )
// grid 256 x 128 threads; one wave = one (n, s-block16, pos-tile16)
__global__ void k_layer2(const float* __restrict__ diag, const float* __restrict__ rows,
                         const float* __restrict__ cols, const float* __restrict__ W0,
                         const float* __restrict__ W2, const float* __restrict__ W3,
                         const float* __restrict__ cst, float* __restrict__ aout,
                         float* __restrict__ bnsum, float* __restrict__ bnsq,
                         float* __restrict__ sumA) {
  const int gw = blockIdx.x * 4 + (threadIdx.x >> 5);
  const int n = gw >> 7, rem = gw & 127;
  const int s0 = (rem >> 5) << 4, i0 = (rem & 31) << 4;
  const int l = threadIdx.x & 31, h = l >> 4, q = l & 15;
  const size_t nb = (size_t)n * 16 * 512;

  v8f acc = {};
  acc = gemm_k16(acc, W0, diag + nb, s0, i0, q, h);
  acc = gemm_k16(acc, W2, rows + nb, s0, i0, q, h);
  acc = gemm_k16(acc, W3, cols + nb, s0, i0, q, h);

#pragma unroll
  for (int r = 0; r < 8; ++r) {
    const int s = s0 + r + 8 * h;
    float v = acc[r] + cst[n * 64 + s];
    v = fmaxf(v, 0.f);
    aout[((size_t)n * 64 + s) * 512 + i0 + q] = v;
    float sv = half16_sum(v);
    float sv2 = half16_sum(v * v);
    if (q == 0) {
      atomicAdd(&bnsum[s], sv);
      atomicAdd(&bnsq[s], sv2);
      atomicAdd(&sumA[n * 64 + s], sv);
    }
  }
}

// ================= K3/K5: BN finalize + fold BN into next 1->1 layer's weights
// A_d = g*rsqrt(var+eps); B_d = beta - A*mu
// Mp[s*64+d] = c[d,s,0]*A_d
// cst[n,s] = bias[s] + sum_d( c[d,s,0]*B_d + c[d,s,1]*(A_d*sumA[n,d]+512*B_d)/512 )
__global__ void k_bn_mix(const float* __restrict__ bnsum, const float* __restrict__ bnsq,
                         const float* __restrict__ g, const float* __restrict__ bt,
                         const float* __restrict__ c, const float* __restrict__ bias,
                         const float* __restrict__ sumA, float* __restrict__ Mp,
                         float* __restrict__ cst) {
  __shared__ float sA[64], sB[64];
  const int tid = threadIdx.x;
  if (tid < 64) {
    float mu = bnsum[tid] * (1.f / 4096.f);
    float var = bnsq[tid] * (1.f / 4096.f) - mu * mu;
    float A = g[tid] * rsqrtf(var + 1e-5f);
    sA[tid] = A;
    sB[tid] = bt[tid] - A * mu;
  }
  __syncthreads();
  for (int idx = tid; idx < 4096; idx += 256) {
    int s = idx >> 6, d = idx & 63;
    Mp[idx] = c[(size_t)(d * 64 + s) * 2] * sA[d];
  }
  for (int idx = tid; idx < 512; idx += 256) {
    int n = idx >> 6, s = idx & 63;
    float acc = bias[s];
    for (int d = 0; d < 64; ++d) {
      float c0 = c[(size_t)(d * 64 + s) * 2];
      float c1 = c[(size_t)(d * 64 + s) * 2 + 1];
      acc += c0 * sB[d] +
             c1 * (sA[d] * sumA[n * 64 + d] + 512.f * sB[d]) * (1.f / 512.f);
    }
    cst[idx] = acc;
  }
}

// =========================== K4/K6: 64x64 WMMA channel mix + ReLU + BN stats
__global__ void k_transform(const float* __restrict__ ain, const float* __restrict__ W,
                            const float* __restrict__ cst, float* __restrict__ aout,
                            float* __restrict__ bnsum, float* __restrict__ bnsq,
                            float* __restrict__ sumA) {
  const int gw = blockIdx.x * 4 + (threadIdx.x >> 5);
  const int n = gw >> 7, rem = gw & 127;
  const int s0 = (rem >> 5) << 4, i0 = (rem & 31) << 4;
  const int l = threadIdx.x & 31, h = l >> 4, q = l & 15;
  const float* S = ain + (size_t)n * 64 * 512;

  v8f acc = {};
#pragma unroll
  for (int k0 = 0; k0 < 64; k0 += 4) {
    const int ka = k0 + 2 * h;
    v2f a, b;
    a.x = W[(s0 + q) * 64 + ka];
    a.y = W[(s0 + q) * 64 + ka + 1];
    b.x = S[ka * 512 + i0 + q];
    b.y = S[(ka + 1) * 512 + i0 + q];
    acc = wmma_f32_k4(a, b, acc);
  }

#pragma unroll
  for (int r = 0; r < 8; ++r) {
    const int s = s0 + r + 8 * h;
    float v = acc[r] + cst[n * 64 + s];
    v = fmaxf(v, 0.f);
    aout[((size_t)n * 64 + s) * 512 + i0 + q] = v;
    float sv = half16_sum(v);
    float sv2 = half16_sum(v * v);
    if (q == 0) {
      atomicAdd(&bnsum[s], sv);
      atomicAdd(&bnsq[s], sv2);
      if (sumA) atomicAdd(&sumA[n * 64 + s], sv);
    }
  }
}

// ===================== K7: BN2 finalize + fold into fc1 weights/bias
__global__ void k_bn_fc(const float* __restrict__ bnsum, const float* __restrict__ bnsq,
                        const float* __restrict__ g, const float* __restrict__ bt,
                        const float* __restrict__ fc1w, const float* __restrict__ fc1b,
                        float* __restrict__ F1, float* __restrict__ cf1) {
  __shared__ float sA[64], sB[64];
  const int tid = threadIdx.x;
  if (tid < 64) {
    float mu = bnsum[tid] * (1.f / 4096.f);
    float var = bnsq[tid] * (1.f / 4096.f) - mu * mu;
    float A = g[tid] * rsqrtf(var + 1e-5f);
    sA[tid] = A;
    sB[tid] = bt[tid] - A * mu;
  }
  __syncthreads();
  for (int idx = tid; idx < 4096; idx += 256) {
    int d = idx & 63;
    F1[idx] = fc1w[idx] * sA[d];
  }
  if (tid < 64) {
    float acc = fc1b[tid];
    for (int d = 0; d < 64; ++d) acc += fc1w[tid * 64 + d] * sB[d];
    cf1[tid] = acc;
  }
}

// ======================= K8: fused fc1(ReLU) + fc2, WMMA, fc1 tile via LDS
// block = 128 thr (4 waves) handles (n, pos-tile16). Waves 0..3 compute 16-row
// slabs of r1 = relu(F1 @ a2 + cf1) into LDS; wave 0 then does 16x16x64 fc2.
__global__ void k_fc(const float* __restrict__ a2, const float* __restrict__ F1,
                     const float* __restrict__ cf1, const float* __restrict__ fc2w,
                     const float* __restrict__ fc2b, float* __restrict__ out) {
  __shared__ float r1[64 * 16];
  const int n = blockIdx.x >> 5, i0 = (blockIdx.x & 31) << 4;
  const int w = threadIdx.x >> 5, l = threadIdx.x & 31, h = l >> 4, q = l & 15;
  const float* S = a2 + (size_t)n * 64 * 512;

  v8f acc = {};
#pragma unroll
  for (int k0 = 0; k0 < 64; k0 += 4) {
    const int ka = k0 + 2 * h;
    v2f a, b;
    a.x = F1[(w * 16 + q) * 64 + ka];
    a.y = F1[(w * 16 + q) * 64 + ka + 1];
    b.x = S[ka * 512 + i0 + q];
    b.y = S[(ka + 1) * 512 + i0 + q];
    acc = wmma_f32_k4(a, b, acc);
  }
#pragma unroll
  for (int r = 0; r < 8; ++r) {
    const int o = w * 16 + r + 8 * h;
    r1[o * 16 + q] = fmaxf(acc[r] + cf1[o], 0.f);
  }
  __syncthreads();

  if (w == 0) {  // wave-uniform branch: EXEC all-ones inside
    v8f acc2 = {};
#pragma unroll
    for (int k0 = 0; k0 < 64; k0 += 4) {
      const int ka = k0 + 2 * h;
      v2f a, b;
      a.x = fc2w[q * 64 + ka];
      a.y = fc2w[q * 64 + ka + 1];
      b.x = r1[ka * 16 + q];
      b.y = r1[(ka + 1) * 16 + q];
      acc2 = wmma_f32_k4(a, b, acc2);
    }
#pragma unroll
    for (int r = 0; r < 8; ++r) {
      const int o = r + 8 * h;  // O = 16 rows exactly
      out[((size_t)n * 16 + o) * 512 + i0 + q] = acc2[r] + fc2b[o];
    }
  }
}

// ============================================================== launcher
extern "C" void kernel_launch(void* const* d_in, const int* in_sizes, int n_in,
                              void* d_out, int out_size, void* d_ws, size_t ws_size,
                              hipStream_t stream) {
  const float* x      = (const float*)d_in[0];
  const float* c2     = (const float*)d_in[1];
  const float* bias2  = (const float*)d_in[2];
  const float* c1a    = (const float*)d_in[3];
  const float* bias1a = (const float*)d_in[4];
  const float* c1b    = (const float*)d_in[5];
  const float* bias1b = (const float*)d_in[6];
  const float* g0 = (const float*)d_in[7],  *b0 = (const float*)d_in[8];
  const float* g1 = (const float*)d_in[9],  *b1 = (const float*)d_in[10];
  const float* g2 = (const float*)d_in[11], *b2 = (const float*)d_in[12];
  const float* fc1w = (const float*)d_in[13], *fc1b = (const float*)d_in[14];
  const float* fc2w = (const float*)d_in[15], *fc2b = (const float*)d_in[16];
  float* out = (float*)d_out;
  float* ws = (float*)d_ws;

  // workspace layout (floats); atomic-accumulated region is contiguous
  float* rows = ws;                 // 65536
  float* diag = ws + 65536;         // 65536
  float* a0   = ws + 131072;        // [8][64][512]
  float* a1   = a0 + 262144;
  float* a2   = a1 + 262144;
  float* zb   = a2 + 262144;        // ---- zeroed-every-call region (67200) ----
  float* cols = zb;                 // 65536 (atomic)
  float* dsum = zb + 65536;         // 128   (atomic)
  float* tot  = zb + 65664;         // 128   (atomic)
  float* st   = zb + 65792;         // 1408  (atomic BN stats)
  float* bnsum0 = st,       *bnsq0 = st + 64;
  float* bnsum1 = st + 128, *bnsq1 = st + 192;
  float* bnsum2 = st + 256, *bnsq2 = st + 320;
  float* sumA0 = st + 384;          // [8][64]
  float* sumA1 = st + 896;          // [8][64]
  float* W0 = zb + 67200, *W2 = W0 + 1024, *W3 = W2 + 1024;  // 64x16 each
  float* cst0 = W3 + 1024;          // [8][64]
  float* M1p = cst0 + 512;          // 64x64
  float* cst1 = M1p + 4096;
  float* M2p = cst1 + 512;
  float* cst2 = M2p + 4096;
  float* F1p = cst2 + 512;
  float* cf1 = F1p + 4096;

  k_zero<<<263, 256, 0, stream>>>(zb, 67200);
  k_reduce_x<<<512, 512, 0, stream>>>(x, rows, cols, diag, dsum, tot);
  k_prep2<<<1, 256, 0, stream>>>(c2, bias2, dsum, tot, W0, W2, W3, cst0);
  k_layer2<<<256, 128, 0, stream>>>(diag, rows, cols, W0, W2, W3, cst0, a0,
                                    bnsum0, bnsq0, sumA0);
  k_bn_mix<<<1, 256, 0, stream>>>(bnsum0, bnsq0, g0, b0, c1a, bias1a, sumA0,
                                  M1p, cst1);
  k_transform<<<256, 128, 0, stream>>>(a0, M1p, cst1, a1, bnsum1, bnsq1, sumA1);
  k_bn_mix<<<1, 256, 0, stream>>>(bnsum1, bnsq1, g1, b1, c1b, bias1b, sumA1,
                                  M2p, cst2);
  k_transform<<<256, 128, 0, stream>>>(a1, M2p, cst2, a2, bnsum2, bnsq2, nullptr);
  k_bn_fc<<<1, 256, 0, stream>>>(bnsum2, bnsq2, g2, b2, fc1w, fc1b, F1p, cf1);
  k_fc<<<256, 128, 0, stream>>>(a2, F1p, cf1, fc2w, fc2b, out);
}